// PairWiseLoss_52501680226569
// MI455X (gfx1250) — compile-verified
//
#include <hip/hip_runtime.h>
#include <hip/hip_bf16.h>

// Pairwise hinge loss, segment-restricted, via V_WMMA_F32_16X16X4_F32.
//
// scores: float32[N], labels: int32[N], segment_ids: int32[N] (sorted, 64 graphs)
// loss = mean over pairs (i,j) with seg[i]==seg[j], lab[i]==1, lab[j]==0 of
//        max(1 - (s_i - s_j), 0);  0 if no valid pair.
//
// WMMA trick: 1 - s_i + s_j is rank-2 => one 16x16x4 f32 WMMA per tile:
//   A row m = [-s_row, 1, 0, 0], B col n = [1, s_col, 0, 0], C = 1
//   D[m][n] = 1 - s_row + s_col  (exact f32)
// Masking + ReLU + exact integer pair count done per D element in VALU.

#define TILE 16
#define WAVES_PER_BLOCK 8
#define BLOCK_THREADS 256

typedef float v2f __attribute__((ext_vector_type(2)));
typedef float v8f __attribute__((ext_vector_type(8)));

// ---------------- kernel 1: segment start offsets via binary search ----------
__global__ void seg_starts_kernel(const int* __restrict__ seg,
                                  int* __restrict__ gstart,
                                  int n, int num_graphs) {
    int g = threadIdx.x + blockIdx.x * blockDim.x;
    if (g > num_graphs) return;                 // g in [0, num_graphs], gstart[num_graphs] = n
    int lo = 0, hi = n;
    while (lo < hi) {
        int mid = (lo + hi) >> 1;
        if (seg[mid] < g) lo = mid + 1; else hi = mid;
    }
    gstart[g] = lo;
}

// ---------------- kernel 2: tiled pairwise hinge with WMMA -------------------
__global__ void pair_hinge_kernel(const float* __restrict__ scores,
                                  const int* __restrict__ labels,
                                  const int* __restrict__ seg,
                                  const int* __restrict__ gstart,
                                  float* __restrict__ psum,
                                  int* __restrict__ pcnt) {
    const int r       = blockIdx.x;            // row tile index
    const int rowbase = r * TILE;
    const int lane    = threadIdx.x & 31;
    const int wave    = threadIdx.x >> 5;
    const int half    = lane >> 4;             // 0: lanes 0-15, 1: lanes 16-31
    const int laneN   = lane & 15;

    // Segment range covered by this row tile (seg is sorted) -> column range.
    const int segA = seg[rowbase];
    const int segB = seg[rowbase + TILE - 1];
    const int c0   = gstart[segA];             // first column with a matching segment
    const int c1   = gstart[segB + 1];         // one past the last such column
    const int j0   = c0 >> 4;
    const int j1   = (c1 - 1) >> 4;            // inclusive; c1 > rowbase+15 >= 15 so safe

    // A operand (16x4 f32): lanes 0-15 hold {K0,K1} = {-s_row, 1}; lanes 16-31 hold {K2,K3}=0.
    const float srow = scores[rowbase + laneN];
    v2f a;
    a.x = half ? 0.0f : -srow;
    a.y = half ? 0.0f : 1.0f;

    // Row metadata for the 8 D elements this lane owns (M = k + 8*half).
    int segrow[8], labrow[8];
#pragma unroll
    for (int k = 0; k < 8; ++k) {
        int rr = rowbase + half * 8 + k;
        segrow[k] = seg[rr];
        labrow[k] = labels[rr];
    }

    float sum = 0.0f;
    int   cnt = 0;

    for (int j = j0 + wave; j <= j1; j += WAVES_PER_BLOCK) {
        const int   col  = j * TILE + laneN;   // this lane's column (N = lane & 15)
        const float scol = scores[col];
        const int   segc = seg[col];
        const int   labc = labels[col];

        // B operand (4x16 f32): lanes 0-15 hold rows {K0,K1} = {1, s_col}; lanes 16-31 rows {K2,K3}=0.
        v2f b;
        b.x = half ? 0.0f : 1.0f;
        b.y = half ? 0.0f : scol;

        v8f cacc = {1.f, 1.f, 1.f, 1.f, 1.f, 1.f, 1.f, 1.f};

        // D[m][n] = 1 - s_row + s_col
        v8f d = __builtin_amdgcn_wmma_f32_16x16x4_f32(
            /*neg_a=*/false, a, /*neg_b=*/false, b,
            /*c_mod=*/(short)0, cacc, /*reuse_a=*/false, /*reuse_b=*/false);

#pragma unroll
        for (int k = 0; k < 8; ++k) {
            bool valid = (segrow[k] == segc) & (labrow[k] == 1) & (labc == 0);
            float h = d[k] > 0.0f ? d[k] : 0.0f;   // hinge
            sum += valid ? h : 0.0f;
            cnt += valid ? 1 : 0;
        }
    }

    // Deterministic wave32 reduction.
#pragma unroll
    for (int off = 16; off > 0; off >>= 1) {
        sum += __shfl_xor(sum, off, 32);
        cnt += __shfl_xor(cnt, off, 32);
    }

    __shared__ float ssum[WAVES_PER_BLOCK];
    __shared__ int   scnt[WAVES_PER_BLOCK];
    if (lane == 0) { ssum[wave] = sum; scnt[wave] = cnt; }
    __syncthreads();
    if (threadIdx.x == 0) {
        float t = 0.0f; int cc = 0;
#pragma unroll
        for (int w = 0; w < WAVES_PER_BLOCK; ++w) { t += ssum[w]; cc += scnt[w]; }
        psum[r] = t;
        pcnt[r] = cc;
    }
}

// ---------------- kernel 3: fixed-order final reduction ----------------------
__global__ void finalize_kernel(const float* __restrict__ psum,
                                const int* __restrict__ pcnt,
                                float* __restrict__ out, int ntiles) {
    __shared__ float fs[BLOCK_THREADS];
    __shared__ int   ic[BLOCK_THREADS];
    const int tid = threadIdx.x;
    float s = 0.0f; int c = 0;
    for (int i = tid; i < ntiles; i += BLOCK_THREADS) { s += psum[i]; c += pcnt[i]; }
    fs[tid] = s; ic[tid] = c;
    __syncthreads();
#pragma unroll
    for (int stride = BLOCK_THREADS / 2; stride > 0; stride >>= 1) {
        if (tid < stride) { fs[tid] += fs[tid + stride]; ic[tid] += ic[tid + stride]; }
        __syncthreads();
    }
    if (tid == 0) out[0] = (ic[0] > 0) ? fs[0] / (float)ic[0] : 0.0f;
}

// ---------------- launch -----------------------------------------------------
extern "C" void kernel_launch(void* const* d_in, const int* in_sizes, int n_in,
                              void* d_out, int out_size, void* d_ws, size_t ws_size,
                              hipStream_t stream) {
    const float* scores = (const float*)d_in[0];
    const int*   labels = (const int*)d_in[1];
    const int*   seg    = (const int*)d_in[2];
    float*       out    = (float*)d_out;

    const int n          = in_sizes[0];        // 12288, divisible by 16
    const int num_graphs = 64;
    const int ntiles     = n / TILE;           // 768

    // Workspace layout: [float psum[ntiles]][int pcnt[ntiles]][int gstart[num_graphs+1]]
    float* psum   = (float*)d_ws;
    int*   pcnt   = (int*)(psum + ntiles);
    int*   gstart = pcnt + ntiles;

    seg_starts_kernel<<<1, 128, 0, stream>>>(seg, gstart, n, num_graphs);
    pair_hinge_kernel<<<ntiles, BLOCK_THREADS, 0, stream>>>(scores, labels, seg,
                                                            gstart, psum, pcnt);
    finalize_kernel<<<1, BLOCK_THREADS, 0, stream>>>(psum, pcnt, out, ntiles);
}